// LocalFeatureLoss_87643102642377
// MI455X (gfx1250) — compile-verified
//
#include <hip/hip_runtime.h>
#include <hip/hip_bf16.h>

// ---------------------------------------------------------------------------
// Problem constants (from reference: xyz (4, 8192, 3) f32, K = 10)
// ---------------------------------------------------------------------------
#define BATCH 4
#define NPTS  8192
#define KNN   10

#define CHUNK 2048              // candidates staged in LDS per pass (32 KB)
#define WAVES_PER_BLOCK 8
#define QUERIES_PER_WAVE 16
#define QUERIES_PER_BLOCK (WAVES_PER_BLOCK * QUERIES_PER_WAVE)   // 128

typedef __attribute__((ext_vector_type(2))) float v2f;
typedef __attribute__((ext_vector_type(8))) float v8f;

// ---------------------------------------------------------------------------
// Sorted top-K insert of 8 WMMA-produced scores (registers only after inline)
// ---------------------------------------------------------------------------
__device__ __forceinline__ void topk_update(float (&dist)[KNN], int (&idx)[KNN],
                                            const v8f& d, int jbase)
{
#pragma unroll
    for (int r = 0; r < 8; ++r) {
        const float s = d[r];
        const int   j = jbase + r;
        if (s < dist[KNN - 1]) {
            dist[KNN - 1] = s;
            idx[KNN - 1]  = j;
#pragma unroll
            for (int u = KNN - 1; u > 0; --u) {
                if (dist[u] < dist[u - 1]) {
                    const float td = dist[u]; dist[u] = dist[u - 1]; dist[u - 1] = td;
                    const int   ti = idx[u];  idx[u]  = idx[u - 1];  idx[u - 1]  = ti;
                }
            }
        }
    }
}

// ---------------------------------------------------------------------------
// Kernel 1: brute-force KNN via V_WMMA_F32_16X16X4_F32 rank-4 GEMM scoring.
//
// score(i,j) = |p_j|^2 - 2 p_i . p_j     (== d_ij - |p_i|^2 : same per-query
//                                          ordering as true squared distance)
//
// Matrix roles per wave (wave32):
//   A (16x4)  = candidate tile:  row m = (x, y, z, |p|^2) of candidate m
//   B (4x16)  = query tile:      col n = (-2x, -2y, -2z, 1) of query n
//   D (16x16) = D[m][n] = score(candidate m, query n)
//
// f32 operand layout (2 VGPRs per lane):
//   A: lanes 0-15 hold M=lane, VGPR r = K=r;  lanes 16-31: VGPR r = K=r+2
//   B: lanes 0-15 hold N=lane, VGPR r = K=r;  lanes 16-31: VGPR r = K=r+2
//   D: lane (n + 16*h), VGPR r  holds D[m = r + 8*h][n]
//
// Lane (n, h) accumulates top-10 over candidates {tile + 8h + r}; lanes n and
// n+16 are merged at the end (disjoint candidate subsets, both sorted).
//
// The A-fragment LDS loads are software-pipelined one iteration ahead so the
// WMMAs consume fragments loaded last iteration: the DS wait at the WMMA is
// then for an *older* outstanding load (s_wait_dscnt 1) that has had a full
// iteration of VALU work to complete, instead of a cold s_wait_dscnt 0.
// ---------------------------------------------------------------------------
__global__ __launch_bounds__(QUERIES_PER_BLOCK * 2)
void knn_kernel(const float* __restrict__ xyz1, int* __restrict__ knn_out)
{
    __shared__ float s_cand[CHUNK * 4];                         // 32 KB
    __shared__ float s_md[WAVES_PER_BLOCK][32][KNN];            // 10 KB
    __shared__ int   s_mi[WAVES_PER_BLOCK][32][KNN];            // 10 KB

    const int tid  = threadIdx.x;
    const int wave = tid >> 5;
    const int lane = tid & 31;
    const int half = lane >> 4;     // 0 or 1
    const int hl   = lane & 15;

    const int blocksPerBatch = NPTS / QUERIES_PER_BLOCK;
    const int b     = blockIdx.x / blocksPerBatch;
    const int qbase = (blockIdx.x % blocksPerBatch) * QUERIES_PER_BLOCK;

    const float* base = xyz1 + (size_t)b * NPTS * 3;

    // -------- B fragment (constant over candidate loop): query n = hl ------
    const int qn = qbase + wave * QUERIES_PER_WAVE + hl;
    const float qx = base[qn * 3 + 0];
    const float qy = base[qn * 3 + 1];
    const float qz = base[qn * 3 + 2];
    const float qaug[4] = { -2.0f * qx, -2.0f * qy, -2.0f * qz, 1.0f };
    v2f bfrag;
    bfrag.x = qaug[half * 2 + 0];
    bfrag.y = qaug[half * 2 + 1];

    // -------- private sorted top-K (ascending score) -----------------------
    float dist[KNN];
    int   idx[KNN];
#pragma unroll
    for (int r = 0; r < KNN; ++r) { dist[r] = 3.4e38f; idx[r] = 0; }

    for (int c0 = 0; c0 < NPTS; c0 += CHUNK) {
        // cooperative stage of candidate chunk into LDS as (x, y, z, |p|^2)
        for (int i = tid; i < CHUNK; i += QUERIES_PER_BLOCK * 2) {
            const float x = base[(c0 + i) * 3 + 0];
            const float y = base[(c0 + i) * 3 + 1];
            const float z = base[(c0 + i) * 3 + 2];
            s_cand[i * 4 + 0] = x;
            s_cand[i * 4 + 1] = y;
            s_cand[i * 4 + 2] = z;
            s_cand[i * 4 + 3] = x * x + y * y + z * z;
        }
        __syncthreads();

        const float* lds_a = &s_cand[hl * 4 + half * 2];

        // prologue: fragments for tiles 0 and 16
        v2f a0, a1;
        a0.x = lds_a[0];       a0.y = lds_a[1];
        a1.x = lds_a[16 * 4];  a1.y = lds_a[16 * 4 + 1];

        for (int t = 0; t < CHUNK; t += 32) {
            // prefetch next iteration's fragments (wraps harmlessly to 0)
            const int tn = (t + 32) & (CHUNK - 1);
            const float* cp0 = lds_a + (size_t)tn * 4;
            const float* cp1 = lds_a + (size_t)(tn + 16) * 4;
            v2f n0, n1;
            n0.x = cp0[0]; n0.y = cp0[1];
            n1.x = cp1[0]; n1.y = cp1[1];

            v8f cz = {};   // fresh accumulator (C = 0) for each tile
            const v8f d0 = __builtin_amdgcn_wmma_f32_16x16x4_f32(
                false, a0, false, bfrag, (short)0, cz, false, false);
            const v8f d1 = __builtin_amdgcn_wmma_f32_16x16x4_f32(
                false, a1, false, bfrag, (short)0, cz, false, false);

            topk_update(dist, idx, d0, c0 + t + half * 8);
            topk_update(dist, idx, d1, c0 + t + 16 + half * 8);

            a0 = n0;
            a1 = n1;
        }
        __syncthreads();
    }

    // -------- merge half-lane partial top-K lists through LDS --------------
#pragma unroll
    for (int r = 0; r < KNN; ++r) {
        s_md[wave][lane][r] = dist[r];
        s_mi[wave][lane][r] = idx[r];
    }
    __syncthreads();

    if (lane < 16) {
        const float* da = &s_md[wave][lane][0];
        const float* db = &s_md[wave][lane + 16][0];
        const int*   xa = &s_mi[wave][lane][0];
        const int*   xb = &s_mi[wave][lane + 16][0];
        int* outp = knn_out + ((size_t)b * NPTS + qn) * KNN;
        int ia = 0, ib = 0;
        for (int r = 0; r < KNN; ++r) {
            const float va = da[ia];
            const float vb = db[ib];
            if (va <= vb) { outp[r] = xa[ia]; ++ia; }
            else          { outp[r] = xb[ib]; ++ib; }
        }
    }
}

// ---------------------------------------------------------------------------
// Smallest eigenvector of a symmetric 3x3 (trigonometric eigenvalues + row
// crosses). Equivalent to the last right-singular vector of the 10x3 patch.
// ---------------------------------------------------------------------------
__device__ __forceinline__ void smallest_evec(
    float a00, float a01, float a02, float a11, float a12, float a22,
    float& nx, float& ny, float& nz)
{
    const float p1 = a01 * a01 + a02 * a02 + a12 * a12;
    const float q  = (a00 + a11 + a22) * (1.0f / 3.0f);
    const float b00 = a00 - q, b11 = a11 - q, b22 = a22 - q;
    const float p2 = b00 * b00 + b11 * b11 + b22 * b22 + 2.0f * p1;
    const float p  = sqrtf(p2 * (1.0f / 6.0f) + 1e-30f);
    const float inv = 1.0f / p;
    const float c00 = b00 * inv, c01 = a01 * inv, c02 = a02 * inv;
    const float c11 = b11 * inv, c12 = a12 * inv, c22 = b22 * inv;
    float r = 0.5f * (c00 * (c11 * c22 - c12 * c12)
                    - c01 * (c01 * c22 - c12 * c02)
                    + c02 * (c01 * c12 - c11 * c02));
    r = fminf(1.0f, fmaxf(-1.0f, r));
    const float phi = acosf(r) * (1.0f / 3.0f);
    // smallest eigenvalue: q + 2 p cos(phi + 2*pi/3)
    const float eig = q + 2.0f * p * cosf(phi + 2.0943951023931953f);

    // rows of (A - eig I)
    const float r0x = a00 - eig, r0y = a01,       r0z = a02;
    const float r1x = a01,       r1y = a11 - eig, r1z = a12;
    const float r2x = a02,       r2y = a12,       r2z = a22 - eig;

    // candidate eigenvectors = pairwise row cross products; keep the largest
    float v0x = r0y * r1z - r0z * r1y, v0y = r0z * r1x - r0x * r1z, v0z = r0x * r1y - r0y * r1x;
    float v1x = r0y * r2z - r0z * r2y, v1y = r0z * r2x - r0x * r2z, v1z = r0x * r2y - r0y * r2x;
    float v2x = r1y * r2z - r1z * r2y, v2y = r1z * r2x - r1x * r2z, v2z = r1x * r2y - r1y * r2x;
    const float n0 = v0x * v0x + v0y * v0y + v0z * v0z;
    const float n1 = v1x * v1x + v1y * v1y + v1z * v1z;
    const float n2 = v2x * v2x + v2y * v2y + v2z * v2z;

    float bx = v0x, by = v0y, bz = v0z, bn = n0;
    if (n1 > bn) { bx = v1x; by = v1y; bz = v1z; bn = n1; }
    if (n2 > bn) { bx = v2x; by = v2y; bz = v2z; bn = n2; }
    if (bn < 1e-20f) { bx = 0.0f; by = 0.0f; bz = 1.0f; bn = 1.0f; }
    const float rn = rsqrtf(bn);
    nx = bx * rn; ny = by * rn; nz = bz * rn;
}

// gather K neighbors of one point from `cloud`, fit plane, return signed
// point-to-plane distance of the point itself
__device__ __forceinline__ float point_to_plane(
    const float* __restrict__ cloud, const int* __restrict__ nbr,
    float px, float py, float pz)
{
    float gx[KNN], gy[KNN], gz[KNN];
    float cx = 0.0f, cy = 0.0f, cz = 0.0f;
#pragma unroll
    for (int r = 0; r < KNN; ++r) {
        const int j = nbr[r];
        gx[r] = cloud[j * 3 + 0];
        gy[r] = cloud[j * 3 + 1];
        gz[r] = cloud[j * 3 + 2];
        cx += gx[r]; cy += gy[r]; cz += gz[r];
    }
    cx *= (1.0f / KNN); cy *= (1.0f / KNN); cz *= (1.0f / KNN);

    float a00 = 0.0f, a01 = 0.0f, a02 = 0.0f, a11 = 0.0f, a12 = 0.0f, a22 = 0.0f;
#pragma unroll
    for (int r = 0; r < KNN; ++r) {
        const float dx = gx[r] - cx, dy = gy[r] - cy, dz = gz[r] - cz;
        a00 += dx * dx; a01 += dx * dy; a02 += dx * dz;
        a11 += dy * dy; a12 += dy * dz; a22 += dz * dz;
    }

    float nx, ny, nz;
    smallest_evec(a00, a01, a02, a11, a12, a22, nx, ny, nz);
    return (px - cx) * nx + (py - cy) * ny + (pz - cz) * nz;
}

// ---------------------------------------------------------------------------
// Kernel 2: per-point plane fits on both clouds + loss terms (atomic sums)
// ---------------------------------------------------------------------------
__global__ __launch_bounds__(256)
void loss_kernel(const float* __restrict__ xyz1, const float* __restrict__ xyz2,
                 const int* __restrict__ knn_idx, float* __restrict__ acc)
{
    const int gid = blockIdx.x * blockDim.x + threadIdx.x;
    const int b = gid / NPTS;
    const int n = gid % NPTS;

    const float* c1 = xyz1 + (size_t)b * NPTS * 3;
    const float* c2 = xyz2 + (size_t)b * NPTS * 3;

    int nbr[KNN];
#pragma unroll
    for (int r = 0; r < KNN; ++r) nbr[r] = knn_idx[(size_t)gid * KNN + r];

    const float p1x = c1[n * 3 + 0], p1y = c1[n * 3 + 1], p1z = c1[n * 3 + 2];
    const float p2x = c2[n * 3 + 0], p2y = c2[n * 3 + 1], p2z = c2[n * 3 + 2];

    const float ptof1 = point_to_plane(c1, nbr, p1x, p1y, p1z);
    const float ptof2 = point_to_plane(c2, nbr, p2x, p2y, p2z);

    const float dpl  = fabsf(ptof1) - fabsf(ptof2);
    float plane = dpl * dpl;
    const float bentv = fmaxf(ptof2 - ptof1, 0.0f);
    float bend  = bentv * bentv;

    // wave32 reduction, then one atomic per wave per accumulator
#pragma unroll
    for (int m = 16; m >= 1; m >>= 1) {
        plane += __shfl_xor(plane, m);
        bend  += __shfl_xor(bend,  m);
    }
    if ((threadIdx.x & 31) == 0) {
        atomicAdd(&acc[0], plane);
        atomicAdd(&acc[1], bend);
    }
}

// ---------------------------------------------------------------------------
// Kernel 3: combine sums into final scalar loss
// ---------------------------------------------------------------------------
__global__ void finalize_kernel(const float* __restrict__ acc, float* __restrict__ out)
{
    const float invn = 1.0f / (float)(BATCH * NPTS);
    out[0] = acc[0] * invn + 5.0f * (acc[1] * invn);
}

// ---------------------------------------------------------------------------
extern "C" void kernel_launch(void* const* d_in, const int* in_sizes, int n_in,
                              void* d_out, int out_size, void* d_ws, size_t ws_size,
                              hipStream_t stream)
{
    const float* xyz1 = (const float*)d_in[0];
    const float* xyz2 = (const float*)d_in[1];
    float* out = (float*)d_out;

    int*   knn = (int*)d_ws;                                      // B*N*K ints
    float* acc = (float*)((char*)d_ws + (size_t)BATCH * NPTS * KNN * sizeof(int));

    hipMemsetAsync(acc, 0, 2 * sizeof(float), stream);

    knn_kernel<<<BATCH * NPTS / QUERIES_PER_BLOCK, QUERIES_PER_BLOCK * 2, 0, stream>>>(xyz1, knn);
    loss_kernel<<<BATCH * NPTS / 256, 256, 0, stream>>>(xyz1, xyz2, knn, acc);
    finalize_kernel<<<1, 1, 0, stream>>>(acc, out);
}